// User_State_Model_60584808677538
// MI455X (gfx1250) — compile-verified
//
#include <hip/hip_runtime.h>
#include <cmath>

typedef float v2f __attribute__((ext_vector_type(2)));
typedef float v8f __attribute__((ext_vector_type(8)));

#define STATE_DIM 128
#define HID_DIM   256
#define MT        64     // rows (batch) per workgroup
#define BLOCK     256    // 8 wave32s
#define KP        32     // K-panel depth staged in LDS

// padded LDS strides (floats): multiple of 4 (16B float4 stores), ==4 mod 64 banks
#define S_STRIDE  132
#define H_STRIDE  260
#define W_STRIDE  34     // even -> 8B-aligned v2f loads

// LDS layout (float offsets)
#define OFF_S    0
#define SZ_S     (MT * S_STRIDE)            // 8448
#define OFF_H1   (OFF_S + SZ_S)
#define SZ_H     (MT * H_STRIDE)            // 16640
#define OFF_H2   (OFF_H1 + SZ_H)
#define OFF_WP0  (OFF_H2 + SZ_H)
#define SZ_WP    (HID_DIM * W_STRIDE)       // 8704
#define OFF_WP1  (OFF_WP0 + SZ_WP)
#define OFF_B1   (OFF_WP1 + SZ_WP)
#define OFF_B2   (OFF_B1 + HID_DIM)
#define OFF_B3   (OFF_B2 + HID_DIM)
#define SMEM_FLOATS (OFF_B3 + STATE_DIM)    // 59776 floats
#define SMEM_BYTES  (SMEM_FLOATS * 4)       // 239104 B < 320KB/WGP

// One MLP layer: dst[MT][N] = act(src[MT][K] @ W[K][N] + bias), or fused
// Euler update dst += dt * (pre-activation) when UPDATE.
// Weights streamed from L2 in KP-deep panels, double-buffered in LDS
// (transposed wp[n][k]); the next panel's global loads are issued before the
// current panel's WMMA loop so L2 latency hides under the matrix ops.
// The panel loop is deliberately NOT unrolled: unrolling it spills the
// staging registers (seen in round 2 as scratch reload + vgpr_msb traffic).
template <int K, int N, bool SILU, bool UPDATE>
__device__ __forceinline__ void layer(const float* __restrict__ gW,
                                      const float* __restrict__ bias,
                                      const float* __restrict__ src, int sstride,
                                      float* __restrict__ dst, int dstride,
                                      float* __restrict__ wp0,
                                      float* __restrict__ wp1, float dt) {
  const int tid   = threadIdx.x;
  const int lane  = tid & 31;
  const int wave  = tid >> 5;
  const int l15   = lane & 15;
  const int hi    = lane >> 4;      // half-wave select
  const int khalf = hi * 2;         // A/B frag: lanes 16-31 hold K+2,K+3
  const int mt    = wave & 3;       // 4 M-tiles, 2 wave-groups over N
  constexpr int NT = (N / 16) / 2;  // N-tiles per wave (8 or 4)
  const int ntBase = (wave >> 2) * NT;

  constexpr int NPANEL = K / KP;
  constexpr int PER = (KP * N / 4) / BLOCK;  // float4s staged per thread (8 or 4)

  v8f acc[NT] = {};
  float4 stage[PER];

  // Prologue: stage panel 0 in registers (the only exposed L2 latency).
  {
    const float4* g4 = (const float4*)gW;
#pragma unroll
    for (int j = 0; j < PER; ++j) stage[j] = g4[tid + j * BLOCK];
  }

#pragma unroll 1  // keep the panel loop rolled: body is already a full panel
  for (int kp = 0; kp < NPANEL; ++kp) {
    float* wp = (kp & 1) ? wp1 : wp0;
    // Transpose-store the staged panel: wp[n][k]
#pragma unroll
    for (int j = 0; j < PER; ++j) {
      int idx4 = tid + j * BLOCK;
      int k = idx4 / (N / 4);
      int n = (idx4 % (N / 4)) * 4;
      wp[(n + 0) * W_STRIDE + k] = stage[j].x;
      wp[(n + 1) * W_STRIDE + k] = stage[j].y;
      wp[(n + 2) * W_STRIDE + k] = stage[j].z;
      wp[(n + 3) * W_STRIDE + k] = stage[j].w;
    }
    __syncthreads();  // single barrier per panel (2-deep buffering makes it safe)

    // Prefetch next panel into registers; these global_load_b128s overlap
    // with the WMMA loop below and are only waited on at the next store.
    if (kp + 1 < NPANEL) {
      const float4* g4 = (const float4*)(gW + (size_t)(kp + 1) * KP * N);
#pragma unroll
      for (int j = 0; j < PER; ++j) stage[j] = g4[tid + j * BLOCK];
    }

    // A frag row pointer: lanes 0-15 -> M=lane (K0,K1); 16-31 -> M=lane-16 (K2,K3)
    const float* arow = src + (mt * 16 + l15) * sstride + kp * KP + khalf;
#pragma unroll
    for (int kk = 0; kk < KP / 4; ++kk) {
      v2f a = *(const v2f*)(arow + kk * 4);
#pragma unroll
      for (int t = 0; t < NT; ++t) {
        // B frag: V0 = rows K0/K2 striped over lanes (N = tile*16 + l15)
        const float* brow =
            wp + ((ntBase + t) * 16 + l15) * W_STRIDE + kk * 4 + khalf;
        v2f b = *(const v2f*)brow;
        acc[t] = __builtin_amdgcn_wmma_f32_16x16x4_f32(
            /*neg_a=*/false, a, /*neg_b=*/false, b,
            /*c_mod=*/(short)0, acc[t], /*reuse_a=*/false, /*reuse_b=*/false);
      }
    }
  }

  // Epilogue in C/D layout: VGPR r -> M = r (+8 for upper half-wave), N = l15
#pragma unroll
  for (int t = 0; t < NT; ++t) {
    int ncol = (ntBase + t) * 16 + l15;
    float bv = bias[ncol];
#pragma unroll
    for (int r = 0; r < 8; ++r) {
      int row = mt * 16 + hi * 8 + r;
      float v = acc[t][r] + bv;
      if (SILU) v = v * __builtin_amdgcn_rcpf(1.0f + __expf(-v));
      if (UPDATE) {
        float* p = dst + row * dstride + ncol;
        *p = *p + dt * v;          // fused Euler: s += dt * f(s)
      } else {
        dst[row * dstride + ncol] = v;
      }
    }
  }
  __syncthreads();  // layer output (and wp buffers) safe for next layer
}

__global__ void __launch_bounds__(BLOCK)
ode_kernel(const float* __restrict__ state,
           const float* __restrict__ W1, const float* __restrict__ b1,
           const float* __restrict__ W2, const float* __restrict__ b2,
           const float* __restrict__ W3, const float* __restrict__ b3,
           float* __restrict__ out, int nsteps, float dtLast) {
  extern __shared__ float smem[];
  float* sS   = smem + OFF_S;
  float* sH1  = smem + OFF_H1;
  float* sH2  = smem + OFF_H2;
  float* sWP0 = smem + OFF_WP0;
  float* sWP1 = smem + OFF_WP1;
  float* sB1  = smem + OFF_B1;
  float* sB2  = smem + OFF_B2;
  float* sB3  = smem + OFF_B3;

  const int tid = threadIdx.x;
  const size_t rowBase = (size_t)blockIdx.x * MT;

  if (tid < HID_DIM) { sB1[tid] = b1[tid]; sB2[tid] = b2[tid]; }
  if (tid < STATE_DIM) sB3[tid] = b3[tid];

  // Load the 64x128 state tile once; it stays LDS-resident for all 41 steps.
  {
    const float4* g4 = (const float4*)(state + rowBase * STATE_DIM);
#pragma unroll
    for (int j = 0; j < (MT * STATE_DIM / 4) / BLOCK; ++j) {  // 8 float4s/thread
      int idx4 = tid + j * BLOCK;
      int r = idx4 / (STATE_DIM / 4);
      int c = (idx4 % (STATE_DIM / 4)) * 4;
      *(float4*)(sS + r * S_STRIDE + c) = g4[idx4];
    }
  }
  __syncthreads();

  for (int it = 0; it < nsteps; ++it) {
    float dt = (it == nsteps - 1) ? dtLast : 0.1f;
    layer<STATE_DIM, HID_DIM, true,  false>(W1, sB1, sS,  S_STRIDE, sH1, H_STRIDE, sWP0, sWP1, 0.0f);
    layer<HID_DIM,   HID_DIM, true,  false>(W2, sB2, sH1, H_STRIDE, sH2, H_STRIDE, sWP0, sWP1, 0.0f);
    layer<HID_DIM, STATE_DIM, false, true >(W3, sB3, sH2, H_STRIDE, sS,  S_STRIDE, sWP0, sWP1, dt);
  }

  // Write final state tile.
  {
    float4* g4 = (float4*)(out + rowBase * STATE_DIM);
#pragma unroll
    for (int j = 0; j < (MT * STATE_DIM / 4) / BLOCK; ++j) {
      int idx4 = tid + j * BLOCK;
      int r = idx4 / (STATE_DIM / 4);
      int c = (idx4 % (STATE_DIM / 4)) * 4;
      g4[idx4] = *(const float4*)(sS + r * S_STRIDE + c);
    }
  }
}

extern "C" void kernel_launch(void* const* d_in, const int* in_sizes, int n_in,
                              void* d_out, int out_size, void* d_ws, size_t ws_size,
                              hipStream_t stream) {
  (void)n_in; (void)out_size; (void)d_ws; (void)ws_size;
  const float* state = (const float*)d_in[0];
  const float* W1 = (const float*)d_in[1];
  const float* b1 = (const float*)d_in[2];
  const float* W2 = (const float*)d_in[3];
  const float* b2 = (const float*)d_in[4];
  const float* W3 = (const float*)d_in[5];
  const float* b3 = (const float*)d_in[6];
  // h == 4 is a trace-time constant in the reference; replicate its Python
  // float math exactly: 40 full steps of float32(0.1), plus a residual step
  // of float32(fmod(4.0, 0.1)) since rest > 1e-5.
  const double H = 4.0, INTERVAL = 0.1;
  int nfull = (int)::floor(H / INTERVAL);          // 40
  double rest = ::fmod(H, INTERVAL);               // ~0.0999999999999999
  int nsteps = nfull + ((rest > 1e-5) ? 1 : 0);    // 41
  float dtLast = (rest > 1e-5) ? (float)rest : 0.1f;

  int rows = in_sizes[0] / STATE_DIM;              // 65536
  dim3 grid(rows / MT), block(BLOCK);

  hipFuncSetAttribute((const void*)ode_kernel,
                      hipFuncAttributeMaxDynamicSharedMemorySize, SMEM_BYTES);
  ode_kernel<<<grid, block, SMEM_BYTES, stream>>>(state, W1, b1, W2, b2, W3, b3,
                                                  (float*)d_out, nsteps, dtLast);
}